// CUDASignalProcessor_3178275799266
// MI455X (gfx1250) — compile-verified
//
#include <hip/hip_runtime.h>

// ---------------------------------------------------------------------------
// Block-parallel state-space IIR for CDNA5 (gfx1250, wave32).
//
// Pipeline (all on `stream`):
//   K0 setup : combine bandpass+notch into one order-10 filter (b,a of 11
//              taps), precompute homogeneous chunk responses H[S][MP].
//   K1 fir   : v = b (*) x                       (parallel, memory-bound)
//   K2 zsr   : per (channel,chunk): zero-state recurrence over S=256 samples,
//              in-place v -> y0, record final-10-state p_c.
//   K3 scan  : per channel: s_{c+1} = p_c + G s_c   (32 sequential 10x10)
//   K4 corr  : y += H @ s  as a batched GEMM using V_WMMA_F32_16X16X4_F32.
// ---------------------------------------------------------------------------

typedef __attribute__((ext_vector_type(2))) float v2f;
typedef __attribute__((ext_vector_type(4))) float v4f;
typedef __attribute__((ext_vector_type(8))) float v8f;

#define NCH     1024          // 16 * 64 independent channels
#define TLEN    8192          // samples per channel
#define SCHUNK  256           // chunk length (>= MC)
#define NCHUNK  (TLEN / SCHUNK)        // 32
#define NI      (NCH * NCHUNK)         // 32768 chunk instances
#define KC      11            // combined numerator taps
#define MC      10            // combined denominator order
#define MP      12            // MC padded to multiple of 4 (WMMA K-steps)

// workspace layout (float offsets); total ~3.1 MB
#define WS_BC   0
#define WS_AC   16
#define WS_H    32                          // SCHUNK * MP
#define WS_P    (32 + SCHUNK * MP)          // NI * MP
#define WS_S    (32 + SCHUNK * MP + NI * MP) // NI * MP

// ---------------------------------------------------------------------------
// K0: combine filters, precompute homogeneous responses H.
// ---------------------------------------------------------------------------
__global__ void iir_setup_kernel(const float* __restrict__ bpb,
                                 const float* __restrict__ bpa,
                                 const float* __restrict__ nb,
                                 const float* __restrict__ na,
                                 float* __restrict__ ws) {
    __shared__ float sac[16];
    int tid = threadIdx.x;
    if (tid == 0) {
        float b1[9], a1[9], b2[3], a2[3];
        float inv1 = 1.0f / bpa[0];
        for (int i = 0; i < 9; ++i) { b1[i] = bpb[i] * inv1; a1[i] = bpa[i] * inv1; }
        float inv2 = 1.0f / na[0];
        for (int i = 0; i < 3; ++i) { b2[i] = nb[i] * inv2; a2[i] = na[i] * inv2; }
        float bc[16], ac[16];
        for (int k = 0; k < 16; ++k) { bc[k] = 0.0f; ac[k] = 0.0f; }
        for (int i = 0; i < 9; ++i)
            for (int j = 0; j < 3; ++j) {
                bc[i + j] += b1[i] * b2[j];
                ac[i + j] += a1[i] * a2[j];
            }
        // ac[0] == 1 (both inputs monic after normalization)
        for (int k = 0; k < 16; ++k) {
            ws[WS_BC + k] = bc[k];
            ws[WS_AC + k] = ac[k];
            sac[k] = ac[k];
        }
    }
    __syncthreads();
    // H column m: response to initial state e_m (y[-1-j] = delta(j==m)), v=0.
    if (tid < MP) {
        int m = tid;
        float* H = ws + WS_H;
        if (m < MC) {
            float at[MC], hist[MC];
            #pragma unroll
            for (int j = 0; j < MC; ++j) at[j] = sac[j + 1];
            #pragma unroll
            for (int j = 0; j < MC; ++j) hist[j] = (j == m) ? 1.0f : 0.0f;
            for (int t = 0; t < SCHUNK; ++t) {
                float v = 0.0f;
                #pragma unroll
                for (int j = 0; j < MC; ++j) v -= at[j] * hist[j];
                H[t * MP + m] = v;
                #pragma unroll
                for (int j = MC - 1; j > 0; --j) hist[j] = hist[j - 1];
                hist[0] = v;
            }
        } else {
            for (int t = 0; t < SCHUNK; ++t) H[t * MP + m] = 0.0f;  // pad cols
        }
    }
}

// ---------------------------------------------------------------------------
// K1: FIR part, v[n] = sum_k b[k] * x[n-k]  (zero history). Parallel.
// ---------------------------------------------------------------------------
__global__ void iir_fir_kernel(const float* __restrict__ x,
                               float* __restrict__ out,
                               const float* __restrict__ ws) {
    int idx = blockIdx.x * blockDim.x + threadIdx.x;   // 0 .. NCH*TLEN-1
    int ch = idx >> 13;                                // / 8192
    int t  = idx & (TLEN - 1);
    const float* b  = ws + WS_BC;
    const float* xc = x + (size_t)ch * TLEN;
    float acc = 0.0f;
    #pragma unroll
    for (int k = 0; k < KC; ++k) {
        float xv = (t - k >= 0) ? xc[t - k] : 0.0f;
        acc += b[k] * xv;
    }
    out[idx] = acc;
}

// ---------------------------------------------------------------------------
// K2: per-(channel,chunk) zero-state recurrence, in place; record p_c.
// ---------------------------------------------------------------------------
__global__ void iir_zsr_kernel(float* __restrict__ y,
                               float* __restrict__ ws) {
    int n = blockIdx.x * blockDim.x + threadIdx.x;     // 0 .. NI-1
    const float* a = ws + WS_AC;
    float at[MC], hist[MC];
    #pragma unroll
    for (int j = 0; j < MC; ++j) at[j] = a[j + 1];
    #pragma unroll
    for (int j = 0; j < MC; ++j) hist[j] = 0.0f;
    float* yp = y + (size_t)n * SCHUNK;                // n*S == ch*T + c*S
    for (int t = 0; t < SCHUNK; ++t) {
        float acc = yp[t];
        #pragma unroll
        for (int j = 0; j < MC; ++j) acc -= at[j] * hist[j];
        yp[t] = acc;
        #pragma unroll
        for (int j = MC - 1; j > 0; --j) hist[j] = hist[j - 1];
        hist[0] = acc;
    }
    float* p = ws + WS_P + (size_t)n * MP;             // p[i] = y0[S-1-i]
    #pragma unroll
    for (int i = 0; i < MC; ++i) p[i] = hist[i];
    p[MC] = 0.0f; p[MC + 1] = 0.0f;                    // pad rows
}

// ---------------------------------------------------------------------------
// K3: per-channel sequential scan over chunks: s_{c+1} = p_c + G s_c,
//     G[i][m] = H[S-1-i][m]. Stores s_c (state at chunk start).
// ---------------------------------------------------------------------------
__global__ void iir_scan_kernel(float* __restrict__ ws) {
    int ch = blockIdx.x * blockDim.x + threadIdx.x;    // 0 .. NCH-1
    const float* H = ws + WS_H;
    const float* P = ws + WS_P;
    float* Sst = ws + WS_S;
    float s[MC];
    #pragma unroll
    for (int i = 0; i < MC; ++i) s[i] = 0.0f;
    for (int c = 0; c < NCHUNK; ++c) {
        size_t n = (size_t)ch * NCHUNK + c;
        float* so = Sst + n * MP;
        #pragma unroll
        for (int i = 0; i < MC; ++i) so[i] = s[i];
        so[MC] = 0.0f; so[MC + 1] = 0.0f;
        float ns[MC];
        const float* p = P + n * MP;
        #pragma unroll
        for (int i = 0; i < MC; ++i) {
            float acc = p[i];
            const float* g = H + (size_t)(SCHUNK - 1 - i) * MP;
            #pragma unroll
            for (int m = 0; m < MC; ++m) acc += g[m] * s[m];
            ns[i] = acc;
        }
        #pragma unroll
        for (int i = 0; i < MC; ++i) s[i] = ns[i];
    }
}

// ---------------------------------------------------------------------------
// K4: correction GEMM with F32 WMMA:
//     Y(S x NI) += H(S x MP) * S(MP x NI), tiled 16(time) x 16(instance),
//     K in 3 steps of 4 via V_WMMA_F32_16X16X4_F32. One wave per tile;
//     uniform control flow so EXEC is all ones.
// ---------------------------------------------------------------------------
__global__ void iir_corr_kernel(float* __restrict__ y,
                                const float* __restrict__ ws) {
    int gtid = blockIdx.x * blockDim.x + threadIdx.x;
    int wave = gtid >> 5;
    int lane = threadIdx.x & 31;
    int it    = wave & (SCHUNK / 16 - 1);   // time-tile within chunk (0..15)
    int ni    = wave >> 4;                  // instance tile (0..2047)
    int t0    = it << 4;
    int inst0 = ni << 4;
    int half  = lane >> 4;                  // 0 or 1
    int lq    = lane & 15;

    const float* H   = ws + WS_H;
    const float* Sst = ws + WS_S;

    // C/D tile: lane handles instance inst0+lq, times t0 + 8*half + [0..7]
    float* cptr = y + (size_t)(inst0 + lq) * SCHUNK + t0 + 8 * half;
    v4f c0 = *(const v4f*)(cptr);
    v4f c1 = *(const v4f*)(cptr + 4);
    v8f c;
    c[0] = c0.x; c[1] = c0.y; c[2] = c0.z; c[3] = c0.w;
    c[4] = c1.x; c[5] = c1.y; c[6] = c1.z; c[7] = c1.w;

    #pragma unroll
    for (int mb = 0; mb < MP; mb += 4) {
        // A (16x4): lane l -> M = l%16, K = mb + 2*(l/16) + {0,1}
        const float* hp = H + (size_t)(t0 + lq) * MP + mb + 2 * half;
        v2f a; a.x = hp[0]; a.y = hp[1];
        // B (4x16): lane l -> N = l%16, K = mb + 2*(l/16) + {0,1}
        const float* sp = Sst + (size_t)(inst0 + lq) * MP + mb + 2 * half;
        v2f b; b.x = sp[0]; b.y = sp[1];
        c = __builtin_amdgcn_wmma_f32_16x16x4_f32(
                /*neg_a=*/false, a, /*neg_b=*/false, b,
                /*c_mod=*/(short)0, c, /*reuse_a=*/false, /*reuse_b=*/false);
    }

    c0.x = c[0]; c0.y = c[1]; c0.z = c[2]; c0.w = c[3];
    c1.x = c[4]; c1.y = c[5]; c1.z = c[6]; c1.w = c[7];
    *(v4f*)(cptr)     = c0;
    *(v4f*)(cptr + 4) = c1;
}

// ---------------------------------------------------------------------------
extern "C" void kernel_launch(void* const* d_in, const int* in_sizes, int n_in,
                              void* d_out, int out_size, void* d_ws, size_t ws_size,
                              hipStream_t stream) {
    const float* x    = (const float*)d_in[0];   // (16,64,8192) f32
    const float* bp_b = (const float*)d_in[1];   // 9
    const float* bp_a = (const float*)d_in[2];   // 9
    const float* n_b  = (const float*)d_in[3];   // 3
    const float* n_a  = (const float*)d_in[4];   // 3
    float* out = (float*)d_out;                  // (16,64,8192) f32
    float* ws  = (float*)d_ws;                   // needs ~3.1 MB

    iir_setup_kernel<<<1, 64, 0, stream>>>(bp_b, bp_a, n_b, n_a, ws);
    iir_fir_kernel<<<(NCH * TLEN) / 256, 256, 0, stream>>>(x, out, ws);
    iir_zsr_kernel<<<NI / 256, 256, 0, stream>>>(out, ws);
    iir_scan_kernel<<<NCH / 256, 256, 0, stream>>>(ws);
    // waves = (S/16) * (NI/16) = 16 * 2048 = 32768 -> 1M threads, 4096 blocks
    iir_corr_kernel<<<(NI / 16) * (SCHUNK / 16) * 32 / 256, 256, 0, stream>>>(out, ws);
}